// PointPillarScatter_37168646980370
// MI455X (gfx1250) — compile-verified
//
#include <hip/hip_runtime.h>
#include <hip/hip_bf16.h>

// PointPillarScatter for MI455X (gfx1250) — single-output-pass gather version.
//
// Instead of zero-fill + scattered RMW stores (which risks ~2x HBM writeback
// due to L2 wr-rinse between the passes), we invert the scatter into a gather:
//   1) map[b][cell] = -1                                (2.25 MB, in d_ws)
//   2) map[b][y*NX+x] = p   for valid pillars           (unique per batch)
//   3) one pass over the output: each thread owns one cell, reads map once,
//      loads the pillar's feature row with b128 loads (or zeros), and emits
//      fully-coalesced stores (wave32 x 4B = one full 128B line per store).
// Output lines are written exactly once -> HBM write traffic = 144 MB, the
// roofline floor (~7.7 us at 23.3 TB/s including the 30 MB feature reads).

#define NUM_BEV   64
#define NX_DIM    704
#define NY_DIM    200
#define GRID_SZ   (NX_DIM * NY_DIM)   // 140800 cells per (batch, channel) plane
#define BATCH_SZ  4

// ---------------------------------------------------------------------------
// Pass 1: fill the cell->pillar map with -1 (b128 stores).
// ---------------------------------------------------------------------------
__global__ __launch_bounds__(256) void ppmap_fill(int4* __restrict__ map4, int n4) {
    int i = blockIdx.x * 256 + threadIdx.x;
    if (i < n4) map4[i] = make_int4(-1, -1, -1, -1);
}

// ---------------------------------------------------------------------------
// Pass 2: populate map[b][flat] = p for valid pillars (unique -> race-free).
// ---------------------------------------------------------------------------
__global__ __launch_bounds__(256) void ppmap_populate(const int* __restrict__ coords,
                                                      const int* __restrict__ npts,
                                                      int*       __restrict__ map,
                                                      int P) {
    int p = blockIdx.x * 256 + threadIdx.x;
    if (p >= P) return;
    if (npts[p] <= 0) return;                       // invalid pillar -> dropped

    const int b    = coords[4 * p + 0];
    const int flat = coords[4 * p + 1] + coords[4 * p + 2] * NX_DIM + coords[4 * p + 3];

    // mode='drop' semantics
    if ((unsigned)flat >= (unsigned)GRID_SZ) return;
    if ((unsigned)b    >= (unsigned)BATCH_SZ) return;

    map[b * GRID_SZ + flat] = p;
}

// ---------------------------------------------------------------------------
// Pass 3: one thread per output cell; write all 64 channels.
//   - map read: 1 scalar load per thread (coalesced).
//   - feature read: 16 masked b128 loads of the contiguous 256B feature row.
//   - stores: for fixed channel, wave32 lanes cover 32 consecutive cells
//     -> one full 128B cacheline per store instruction, written exactly once.
// ---------------------------------------------------------------------------
__global__ __launch_bounds__(256) void ppgather_write(const float* __restrict__ feat,
                                                      const int*   __restrict__ map,
                                                      float*       __restrict__ out) {
    const int cell = blockIdx.x * 256 + threadIdx.x;   // 0 .. GRID_SZ-1
    const int b    = blockIdx.y;                       // 0 .. BATCH_SZ-1
    if (cell >= GRID_SZ) return;

    const int p = map[b * GRID_SZ + cell];
    const float* frow = feat + (size_t)(p < 0 ? 0 : p) * NUM_BEV;

    if (p >= 0) {
        // Pull the 256B feature row toward the WGP (global_prefetch_b8).
        __builtin_prefetch(frow, 0, 0);
        __builtin_prefetch(frow + 32, 0, 0);
    }

    float* orow = out + (size_t)b * NUM_BEV * GRID_SZ + cell;

#pragma unroll 4
    for (int c = 0; c < NUM_BEV; c += 4) {
        float4 f = make_float4(0.0f, 0.0f, 0.0f, 0.0f);
        if (p >= 0) f = *(const float4*)(frow + c);    // exec-masked b128 load
        orow[(size_t)(c + 0) * GRID_SZ] = f.x;
        orow[(size_t)(c + 1) * GRID_SZ] = f.y;
        orow[(size_t)(c + 2) * GRID_SZ] = f.z;
        orow[(size_t)(c + 3) * GRID_SZ] = f.w;
    }
}

// ---------------------------------------------------------------------------
// Fallback (only if d_ws is too small for the map): zero-fill + scatter.
// ---------------------------------------------------------------------------
__global__ __launch_bounds__(256) void ppscatter_zero_fill(float4* __restrict__ out4,
                                                           int n4,
                                                           float* __restrict__ out,
                                                           int total) {
    int i = blockIdx.x * 256 + threadIdx.x;
    if (i < n4) out4[i] = make_float4(0.0f, 0.0f, 0.0f, 0.0f);
    if (i == 0) {
        for (int r = n4 * 4; r < total; ++r) out[r] = 0.0f;
    }
}

__global__ __launch_bounds__(256) void ppscatter_scatter(const float* __restrict__ feat,
                                                         const int*   __restrict__ coords,
                                                         const int*   __restrict__ npts,
                                                         float*       __restrict__ out,
                                                         int P) {
    int t = blockIdx.x * 256 + threadIdx.x;
    int p = t >> 6;
    int c = t & 63;
    if (p >= P) return;
    if (npts[p] <= 0) return;

    const int b    = coords[4 * p + 0];
    const int flat = coords[4 * p + 1] + coords[4 * p + 2] * NX_DIM + coords[4 * p + 3];
    if ((unsigned)flat >= (unsigned)GRID_SZ) return;
    if ((unsigned)b    >= (unsigned)BATCH_SZ) return;

    out[((size_t)b * NUM_BEV + c) * GRID_SZ + flat] = feat[(size_t)p * NUM_BEV + c];
}

// ---------------------------------------------------------------------------
// Host-side launcher.
//   d_in[0] = pillar_features  (P x 64, f32)
//   d_in[1] = voxel_coords     (P x 4, int32 on device)
//   d_in[2] = voxel_num_points (P, int32)
//   d_in[3] = record_len       (unused)
//   d_out   = (4, 64, 200, 704) f32
// ---------------------------------------------------------------------------
extern "C" void kernel_launch(void* const* d_in, const int* in_sizes, int n_in,
                              void* d_out, int out_size, void* d_ws, size_t ws_size,
                              hipStream_t stream) {
    const float* feat   = (const float*)d_in[0];
    const int*   coords = (const int*)  d_in[1];
    const int*   npts   = (const int*)  d_in[2];
    float*       out    = (float*)d_out;

    const int P = in_sizes[2];   // number of pillars (120000)

    const size_t map_bytes = (size_t)BATCH_SZ * GRID_SZ * sizeof(int);  // 2.25 MB

    if (ws_size >= map_bytes) {
        int* map = (int*)d_ws;

        // Pass 1: map = -1
        const int n4 = (BATCH_SZ * GRID_SZ) / 4;               // 140800 int4s
        ppmap_fill<<<(n4 + 255) / 256, 256, 0, stream>>>((int4*)map, n4);

        // Pass 2: map[b][flat] = p
        ppmap_populate<<<(P + 255) / 256, 256, 0, stream>>>(coords, npts, map, P);

        // Pass 3: single coalesced pass over the 144 MB output.
        dim3 grid((GRID_SZ + 255) / 256, BATCH_SZ);            // (550, 4)
        ppgather_write<<<grid, 256, 0, stream>>>(feat, map, out);
    } else {
        // Fallback: zero-fill + scatter (deterministic choice by ws_size only).
        const int n4 = out_size >> 2;
        ppscatter_zero_fill<<<(n4 + 255) / 256, 256, 0, stream>>>((float4*)out, n4, out, out_size);

        long long threads = (long long)P * NUM_BEV;
        int blocks = (int)((threads + 255) / 256);
        ppscatter_scatter<<<blocks, 256, 0, stream>>>(feat, coords, npts, out, P);
    }
}